// PositionCollapseLoss_35338990911616
// MI455X (gfx1250) — compile-verified
//
#include <hip/hip_runtime.h>
#include <hip/hip_bf16.h>
#include <math.h>

// ---------------------------------------------------------------------------
// PositionCollapseLoss for MI455X (gfx1250, wave32)
//
// Phase 1: streaming second-moment accumulation with V_WMMA_F32_16X16X4_F32.
//   Homogeneous trick: per point v=(x,y,z,1); acc += v v^T gives all second
//   moments + sums + count in one 4x4 block of the 16x16 WMMA accumulator.
//   With the CDNA5 ISA lane layouts, A(16x4) and B(4x16) for this symmetric
//   product are the SAME register pair.
// Phase 2: closed-form eigvalsh of the 3x3 covariance, ratio/log/mean.
// ---------------------------------------------------------------------------

typedef __attribute__((ext_vector_type(2))) float v2f;
typedef __attribute__((ext_vector_type(8))) float v8f;

#define NPTS            65536
#define NFEAT           14
#define BLKS_PER_BATCH  32
#define PTS_PER_BLOCK   (NPTS / BLKS_PER_BATCH)        // 2048
#define WAVES_PER_BLOCK 8
#define PTS_PER_WAVE    (PTS_PER_BLOCK / WAVES_PER_BLOCK) // 256
#define ITERS           (PTS_PER_WAVE / 32)            // 8
#define EPS             1e-6f
#define MAX_RATIO       1000000.0f
#define CLAMP_ABS       1000000.0f

__global__ void zero_ws_kernel(float* ws, int n) {
    int i = blockIdx.x * blockDim.x + threadIdx.x;
    if (i < n) ws[i] = 0.0f;
}

__global__ __launch_bounds__(256)
void cov_wmma_kernel(const float* __restrict__ g, float* __restrict__ ws) {
    // Staging: 8 waves * 32 points * 16 floats (64B/slot, bytes 16..63 stay 0)
    __shared__ float sh[WAVES_PER_BLOCK * 32 * 16];

    const int tid  = threadIdx.x;
    const int lane = tid & 31;
    const int w    = tid >> 5;
    const int b     = blockIdx.x / BLKS_PER_BATCH;
    const int chunk = blockIdx.x % BLKS_PER_BATCH;

    // Zero the staging region once; later writes only touch bytes 0..15 of
    // each 64B slot, so rows m=4..15 of the WMMA A-matrix stay zero forever.
    float4* sh4 = (float4*)sh;
    #pragma unroll
    for (int i = 0; i < (WAVES_PER_BLOCK * 32 * 16 / 4) / 256; ++i)
        sh4[tid + i * 256] = make_float4(0.f, 0.f, 0.f, 0.f);
    __syncthreads();

    const int    waveBase  = w * 512;           // floats
    const int    m         = lane & 15;
    const int    hi        = lane >> 4;
    const size_t batchBase = (size_t)b * NPTS * NFEAT;
    const int    p0        = chunk * PTS_PER_BLOCK + w * PTS_PER_WAVE;

    v8f acc = {};

    for (int it = 0; it < ITERS; ++it) {
        const int p = p0 + it * 32 + lane;
        const float* src = g + batchBase + (size_t)p * NFEAT;
        // 12 contiguous bytes (x,y,z): wave sweeps memory densely at stride 56B
        float x = src[0];
        float y = src[1];
        float z = src[2];
        if (it + 1 < ITERS)
            __builtin_prefetch(src + 32 * NFEAT, 0, 0);   // global_prefetch_b8

        // Stage (x,y,z,1) — homogeneous coordinate carries sums & count.
        *(float4*)&sh[waveBase + lane * 16] = make_float4(x, y, z, 1.0f);
        // same-wave LDS ops are in-order (DScnt); no barrier needed.

        // 8 WMMAs x 4 points = 32 points. A(16x4): lane m holds coord_m of
        // points {2*hi, 2*hi+1} of the quad; identical registers serve as B.
        #pragma unroll
        for (int gq = 0; gq < 8; ++gq) {
            const int slot = gq * 4 + 2 * hi;
            v2f a;
            a[0] = sh[waveBase + slot * 16 + m];        // point 2*hi
            a[1] = sh[waveBase + slot * 16 + 16 + m];   // point 2*hi+1
            acc = __builtin_amdgcn_wmma_f32_16x16x4_f32(
                false, a, false, a, (short)0, acc, false, false);
        }
    }

    // D[m][n] lives in VGPR m (m<8), lane n (n<16). We need the 4x4 block:
    // lane n in 0..3 holds column n in acc[0..3]. Atomically fold into ws.
    if (lane < 4) {
        #pragma unroll
        for (int r = 0; r < 4; ++r)
            unsafeAtomicAdd(&ws[b * 16 + r * 4 + lane], acc[r]);
    }
}

__device__ __forceinline__ float sanitize1(float v) {
    v = isnan(v) ? 0.0f : v;                       // nan -> 0
    return fminf(fmaxf(v, -CLAMP_ABS), CLAMP_ABS); // +/-inf clamp, clip
}

__global__ __launch_bounds__(32)
void eig_finalize_kernel(const float* __restrict__ ws, float* __restrict__ out,
                         int B, float invN) {
    __shared__ float red[32];
    const int t = threadIdx.x;
    float val = 0.0f;

    if (t < B) {
        const int o = t * 16;
        // second moments M[i][j] and sums M[3][i]
        float mu0 = ws[o + 12] * invN;
        float mu1 = ws[o + 13] * invN;
        float mu2 = ws[o + 14] * invN;

        float c00 = sanitize1(ws[o + 0] * invN - mu0 * mu0);
        float c01 = sanitize1(ws[o + 1] * invN - mu0 * mu1);
        float c02 = sanitize1(ws[o + 2] * invN - mu0 * mu2);
        float c10 = sanitize1(ws[o + 4] * invN - mu1 * mu0);
        float c11 = sanitize1(ws[o + 5] * invN - mu1 * mu1);
        float c12 = sanitize1(ws[o + 6] * invN - mu1 * mu2);
        float c20 = sanitize1(ws[o + 8] * invN - mu2 * mu0);
        float c21 = sanitize1(ws[o + 9] * invN - mu2 * mu1);
        float c22 = sanitize1(ws[o + 10] * invN - mu2 * mu2);

        // symmetrize
        float a00 = c00, a11 = c11, a22 = c22;
        float a01 = 0.5f * (c01 + c10);
        float a02 = 0.5f * (c02 + c20);
        float a12 = 0.5f * (c12 + c21);

        // closed-form eigvalsh of symmetric 3x3 (trigonometric method)
        float q  = (a00 + a11 + a22) * (1.0f / 3.0f);
        float b00 = a00 - q, b11 = a11 - q, b22 = a22 - q;
        float p2 = b00 * b00 + b11 * b11 + b22 * b22
                 + 2.0f * (a01 * a01 + a02 * a02 + a12 * a12);
        float p  = sqrtf(p2 * (1.0f / 6.0f));

        float emax, emin;
        if (p < 1e-20f) {
            emax = q; emin = q;
        } else {
            float ip  = 1.0f / p;
            float d00 = b00 * ip, d11 = b11 * ip, d22 = b22 * ip;
            float d01 = a01 * ip, d02 = a02 * ip, d12 = a12 * ip;
            float det = d00 * (d11 * d22 - d12 * d12)
                      - d01 * (d01 * d22 - d12 * d02)
                      + d02 * (d01 * d12 - d11 * d02);
            float r   = fminf(fmaxf(0.5f * det, -1.0f), 1.0f);
            float phi = acosf(r) * (1.0f / 3.0f);
            emax = q + 2.0f * p * cosf(phi);
            emin = q + 2.0f * p * cosf(phi + 2.0943951023931953f); // +2pi/3
        }

        emax = fmaxf(sanitize1(emax), EPS);
        emin = fmaxf(sanitize1(emin), EPS);
        float ratio = fminf(fmaxf(emax / emin, 1.0f), MAX_RATIO);
        val = logf(ratio);
    }

    red[t] = val;
    __syncthreads();
    if (t == 0) {
        float s = 0.0f;
        #pragma unroll
        for (int i = 0; i < 32; ++i) s += red[i];
        out[0] = -s / (float)B;
    }
}

extern "C" void kernel_launch(void* const* d_in, const int* in_sizes, int n_in,
                              void* d_out, int out_size, void* d_ws, size_t ws_size,
                              hipStream_t stream) {
    const float* g   = (const float*)d_in[0];
    float*       out = (float*)d_out;
    float*       ws  = (float*)d_ws;

    const int total = in_sizes[0];
    const int B     = total / (NPTS * NFEAT);   // 32 for the reference shapes

    zero_ws_kernel<<<1, 512, 0, stream>>>(ws, B * 16);
    cov_wmma_kernel<<<B * BLKS_PER_BATCH, 256, 0, stream>>>(g, ws);
    eig_finalize_kernel<<<1, 32, 0, stream>>>(ws, out, B, 1.0f / (float)NPTS);
}